// get_model_90752658965071
// MI455X (gfx1250) — compile-verified
//
#include <hip/hip_runtime.h>

#define BN_INV 0.9999950000374997f   // 1/sqrt(1 + 1e-5)

typedef __attribute__((ext_vector_type(2))) float v2f;
typedef __attribute__((ext_vector_type(8))) float v8f;

// ---------------------------------------------------------------------------
// transpose (B,3,N) -> (B,N,3)
// ---------------------------------------------------------------------------
__global__ void transpose_xyz_kernel(const float* __restrict__ in,
                                     float* __restrict__ out, int B, int N) {
    int t = blockIdx.x * blockDim.x + threadIdx.x;
    if (t >= B * N) return;
    int b = t / N, n = t % N;
    const float* ib = in + (size_t)b * 3 * N;
    float* ob = out + (size_t)t * 3;
    ob[0] = ib[n];
    ob[1] = ib[N + n];
    ob[2] = ib[2 * N + n];
}

// ---------------------------------------------------------------------------
// Pack W (Cout, Cin) into K-pair-interleaved layout:
//   element (c, o) -> Wt2[(c>>1)*2*Cout + 2*o + (c&1)]
// so a B-fragment lane (rows c, c+1 at one column) is one contiguous float2.
// ---------------------------------------------------------------------------
__global__ void pack_w_kernel(const float* __restrict__ W,
                              float* __restrict__ Wt2, int Cout, int Cin) {
    int t = blockIdx.x * blockDim.x + threadIdx.x;
    if (t >= Cout * Cin) return;
    int o = t / Cin, c = t % Cin;
    Wt2[(size_t)(c >> 1) * 2 * Cout + 2 * o + (c & 1)] = W[t];
}

// ---------------------------------------------------------------------------
// Farthest point sampling: one block per batch. dist[] lives in LDS.
// ---------------------------------------------------------------------------
__global__ void fps_kernel(const float* __restrict__ xyz, int N, int npoint,
                           int* __restrict__ out) {
    extern __shared__ unsigned char smemraw[];
    float* dist = (float*)smemraw;
    float* rv   = dist + N;
    int*   ri   = (int*)(rv + blockDim.x);

    int b = blockIdx.x, tid = threadIdx.x, bs = blockDim.x;
    const float* xb = xyz + (size_t)b * N * 3;

    for (int i = tid; i < N; i += bs) dist[i] = 1e10f;
    __syncthreads();

    int far = 0;
    for (int t = 0; t < npoint; ++t) {
        if (tid == 0) out[(size_t)b * npoint + t] = far;
        float cx = xb[far * 3 + 0], cy = xb[far * 3 + 1], cz = xb[far * 3 + 2];
        float best = -1.0f; int bi = 0;
        for (int i = tid; i < N; i += bs) {
            float dx = xb[i * 3 + 0] - cx;
            float dy = xb[i * 3 + 1] - cy;
            float dz = xb[i * 3 + 2] - cz;
            float d  = dx * dx + dy * dy + dz * dz;
            float nd = fminf(dist[i], d);
            dist[i] = nd;
            if (nd > best) { best = nd; bi = i; }
        }
        rv[tid] = best; ri[tid] = bi;
        __syncthreads();
        for (int s = bs >> 1; s > 0; s >>= 1) {
            if (tid < s) {
                float ov = rv[tid + s]; int oi = ri[tid + s];
                if (ov > rv[tid] || (ov == rv[tid] && oi < ri[tid])) {
                    rv[tid] = ov; ri[tid] = oi;
                }
            }
            __syncthreads();
        }
        far = ri[0];
        __syncthreads();
    }
}

// ---------------------------------------------------------------------------
// gather centroids: out[b,s,:] = xyz[b, idx[b,s], :]
// ---------------------------------------------------------------------------
__global__ void gather_xyz_kernel(const float* __restrict__ xyz,
                                  const int* __restrict__ idx,
                                  float* __restrict__ out,
                                  int N, int S, int BS) {
    int t = blockIdx.x * blockDim.x + threadIdx.x;
    if (t >= BS) return;
    int b = t / S;
    int j = idx[t];
    const float* src = xyz + ((size_t)b * N + j) * 3;
    out[(size_t)t * 3 + 0] = src[0];
    out[(size_t)t * 3 + 1] = src[1];
    out[(size_t)t * 3 + 2] = src[2];
}

// ---------------------------------------------------------------------------
// Ball query: first K indices (ascending) with d2 <= r2, padded with first.
// ---------------------------------------------------------------------------
__global__ void ball_query_kernel(const float* __restrict__ xyz,
                                  const float* __restrict__ nxyz,
                                  int N, int S, int K, float r2,
                                  int* __restrict__ out, int BS) {
    int t = blockIdx.x * blockDim.x + threadIdx.x;
    if (t >= BS) return;
    int b = t / S;
    const float* xb = xyz + (size_t)b * N * 3;
    float cx = nxyz[(size_t)t * 3 + 0];
    float cy = nxyz[(size_t)t * 3 + 1];
    float cz = nxyz[(size_t)t * 3 + 2];
    int cnt = 0, first = 0;
    for (int j = 0; j < N && cnt < K; ++j) {
        float dx = xb[j * 3 + 0] - cx;
        float dy = xb[j * 3 + 1] - cy;
        float dz = xb[j * 3 + 2] - cz;
        if (dx * dx + dy * dy + dz * dz <= r2) {
            if (cnt == 0) first = j;
            out[(size_t)t * K + cnt] = j;
            ++cnt;
        }
    }
    for (; cnt < K; ++cnt) out[(size_t)t * K + cnt] = first;
}

// ---------------------------------------------------------------------------
// Generic MLP layer via V_WMMA_F32_16X16X4_F32.
// One wave owns a 16 x (16*NT) output tile; NT is compile-time so the hot
// loop has NO conditional accumulation. Software pipelined one K-step deep.
// All hot-loop loads are contiguous float2 (merge to global_load_b64) and
// are indexed as integer offsets off kernel args (global addr space).
//
// mode 1 (first layer of a branch) fuses the ball-query gather:
//   channels [0, Cpts)      -> points[b, idx, c]          (Cpts % 4 == 0)
//   channels [Cpts, Cpts+3) -> xyz[b, idx, :] - new_xyz   (closed-form tail)
// ---------------------------------------------------------------------------
template <int NT>
__global__ __launch_bounds__(32) void mlp_gemm_kernel(
    const float* __restrict__ X,       // mode 0 input (P, Cin)
    const float* __restrict__ points,  // mode 1: (B, Npts, Cpts) or null
    const float* __restrict__ xyz,     // mode 1: (B, Npts, 3)
    const float* __restrict__ newxyz,  // mode 1: (B, S, 3)
    const int*   __restrict__ bidx,    // mode 1: (B, S, K)
    int Npts, int S, int K, int Cpts,
    const float* __restrict__ Wt2,     // K-pair-interleaved weights
    const float* __restrict__ bias,
    const float* __restrict__ gamma, const float* __restrict__ beta,
    float* __restrict__ Y, int P, int Cin, int Cout,
    int ldY, int colOff, int mode) {
    int lane = threadIdx.x;
    int m0 = blockIdx.x * 16;
    int n0 = blockIdx.y * (16 * NT);
    int hi  = lane >> 4;     // 0: lanes 0-15 (K0,K1), 1: lanes 16-31 (K2,K3)
    int l15 = lane & 15;
    int rA  = m0 + l15;

    int tc[NT];   // output column per tile
    int tcp[NT];  // packed-weight column offset (2 floats per column)
#pragma unroll
    for (int t = 0; t < NT; ++t) {
        tc[t]  = n0 + 16 * t + l15;
        tcp[t] = 2 * tc[t];
    }

    // Per-lane A-row context (computed once).
    size_t aOff = 0;          // float index into the A source
    const float* aSrc = X;
    float dx = 0.f, dy = 0.f, dz = 0.f;
    if (mode == 0) {
        aOff = (size_t)rA * Cin;
    } else {
        int k  = rA % K;
        int rs = rA / K;
        int s  = rs % S;
        int b  = rs / S;
        int j  = bidx[(size_t)(b * S + s) * K + k];
        size_t px = ((size_t)b * Npts + j) * 3;
        size_t pn = (size_t)(b * S + s) * 3;
        dx = xyz[px + 0] - newxyz[pn + 0];
        dy = xyz[px + 1] - newxyz[pn + 1];
        dz = xyz[px + 2] - newxyz[pn + 2];
        aSrc = points;
        aOff = points ? ((size_t)b * Npts + j) * Cpts : 0;
    }

    int CinMain = (mode == 0) ? (Cin & ~3) : Cpts;
    int steps   = CinMain >> 2;

    v8f acc[NT];
#pragma unroll
    for (int t = 0; t < NT; ++t)
        acc[t] = (v8f){0.f, 0.f, 0.f, 0.f, 0.f, 0.f, 0.f, 0.f};

    // ---- main loop: unguarded, software pipelined, all-b64 loads ----
    if (steps > 0) {
        size_t ai = aOff + hi * 2;          // A: floats c0, c0+1 (contiguous)
        int    wi = hi * 2 * Cout;          // packed W row-pair base
        const int wStep = 4 * Cout;

        v2f a, bb[NT];
        a.x = aSrc[ai];
        a.y = aSrc[ai + 1];
#pragma unroll
        for (int t = 0; t < NT; ++t) {
            bb[t].x = Wt2[wi + tcp[t]];
            bb[t].y = Wt2[wi + tcp[t] + 1];
        }

#pragma unroll 2
        for (int it = 1; it < steps; ++it) {
            ai += 4;
            wi += wStep;
            v2f an;
            an.x = aSrc[ai];
            an.y = aSrc[ai + 1];
            v2f bn[NT];
#pragma unroll
            for (int t = 0; t < NT; ++t) {
                bn[t].x = Wt2[wi + tcp[t]];
                bn[t].y = Wt2[wi + tcp[t] + 1];
            }
#pragma unroll
            for (int t = 0; t < NT; ++t)
                acc[t] = __builtin_amdgcn_wmma_f32_16x16x4_f32(
                    false, a, false, bb[t], (short)0, acc[t], false, false);
            a = an;
#pragma unroll
            for (int t = 0; t < NT; ++t) bb[t] = bn[t];
        }
#pragma unroll
        for (int t = 0; t < NT; ++t)
            acc[t] = __builtin_amdgcn_wmma_f32_16x16x4_f32(
                false, a, false, bb[t], (short)0, acc[t], false, false);
    }

    // ---- tail (executed once; guarded scalar loads are fine here) ----
    if (mode == 1) {
        // 3 xyz-diff channels at rows [Cpts, Cpts+3); K-slot 3 is zero.
        v2f a;
        a.x = hi ? dz : dx;
        a.y = hi ? 0.f : dy;
        int c0 = Cpts + hi * 2;            // even (Cpts is even)
        size_t wrow = (size_t)(c0 >> 1) * 2 * Cout;
#pragma unroll
        for (int t = 0; t < NT; ++t) {
            v2f bb;
            bb.x = Wt2[wrow + tcp[t]];
            bb.y = hi ? 0.f : Wt2[wrow + tcp[t] + 1];  // row Cpts+3 == Cin
            acc[t] = __builtin_amdgcn_wmma_f32_16x16x4_f32(
                false, a, false, bb, (short)0, acc[t], false, false);
        }
    } else if (Cin & 3) {
        int c0 = CinMain + hi * 2;
        v2f a;
        a.x = (c0     < Cin) ? aSrc[aOff + c0]     : 0.f;
        a.y = (c0 + 1 < Cin) ? aSrc[aOff + c0 + 1] : 0.f;
#pragma unroll
        for (int t = 0; t < NT; ++t) {
            v2f bb;
            bb.x = (c0 < Cin)
                 ? Wt2[(size_t)(c0 >> 1) * 2 * Cout + tcp[t] + (c0 & 1)] : 0.f;
            bb.y = (c0 + 1 < Cin)
                 ? Wt2[(size_t)((c0 + 1) >> 1) * 2 * Cout + tcp[t] + ((c0 + 1) & 1)]
                 : 0.f;
            acc[t] = __builtin_amdgcn_wmma_f32_16x16x4_f32(
                false, a, false, bb, (short)0, acc[t], false, false);
        }
    }

    // ---- epilogue: bias + folded BN + ReLU ----
#pragma unroll
    for (int t = 0; t < NT; ++t) {
        int col  = tc[t];
        float sc = gamma[col] * BN_INV;
        float bo = bias[col];
        float bt = beta[col];
#pragma unroll
        for (int r = 0; r < 8; ++r) {
            int m = m0 + r + hi * 8;
            float v = (acc[t][r] + bo) * sc + bt;
            Y[(size_t)m * ldY + colOff + col] = v > 0.f ? v : 0.f;
        }
    }
}

// ---------------------------------------------------------------------------
// max-pool over K rows per group, writing into a concat slice
// ---------------------------------------------------------------------------
__global__ void maxpool_kernel(const float* __restrict__ feat,
                               float* __restrict__ out,
                               int G, int K, int C, int ldOut, int colOff) {
    int t = blockIdx.x * blockDim.x + threadIdx.x;
    if (t >= G * C) return;
    int g = t / C, c = t % C;
    float m = -3.4e38f;
    for (int k = 0; k < K; ++k)
        m = fmaxf(m, feat[((size_t)g * K + k) * C + c]);
    out[(size_t)g * ldOut + colOff + c] = m;
}

// ---------------------------------------------------------------------------
// build SA3 input: (2048, 643) = concat(l2_xyz(.,3), l2_pts(.,640))
// ---------------------------------------------------------------------------
__global__ void concat_sa3_kernel(const float* __restrict__ xyzp,
                                  const float* __restrict__ pts,
                                  float* __restrict__ out, int R) {
    int t = blockIdx.x * blockDim.x + threadIdx.x;
    if (t >= R * 643) return;
    int r = t / 643, c = t % 643;
    out[t] = (c < 3) ? xyzp[(size_t)r * 3 + c]
                     : pts[(size_t)r * 640 + (c - 3)];
}

// ---------------------------------------------------------------------------
extern "C" void kernel_launch(void* const* d_in, const int* in_sizes, int n_in,
                              void* d_out, int out_size, void* d_ws,
                              size_t ws_size, hipStream_t stream) {
    (void)in_sizes; (void)n_in; (void)out_size; (void)ws_size;
    const int B = 32, N = 8192;
    const float* xyz_in = (const float*)d_in[0];

    char* base = (char*)d_ws;
    auto alloc = [&](size_t bytes) -> char* {
        char* p = base;
        base += (bytes + 255) & ~(size_t)255;
        return p;
    };
    float* xyz_t  = (float*)alloc((size_t)B * N * 3 * 4);
    int*   fps1   = (int*)  alloc((size_t)B * 128 * 4);
    float* l1_xyz = (float*)alloc((size_t)B * 128 * 3 * 4);
    float* l1_pts = (float*)alloc((size_t)B * 128 * 320 * 4);
    int*   fps2   = (int*)  alloc((size_t)B * 64 * 4);
    float* l2_xyz = (float*)alloc((size_t)B * 64 * 3 * 4);
    float* l2_pts = (float*)alloc((size_t)B * 64 * 640 * 4);
    float* cat3   = (float*)alloc((size_t)2048 * 643 * 4);
    int*   ballIdx= (int*)  alloc((size_t)B * 128 * 16 * 4);
    float* wtBuf  = (float*)alloc(((size_t)1024 * 512 + 2048) * 4);
    float* tmp0   = (float*)alloc((size_t)8388608 * 4);
    float* tmp1   = (float*)alloc((size_t)8388608 * 4);

    transpose_xyz_kernel<<<(B * N + 255) / 256, 256, 0, stream>>>(
        xyz_in, xyz_t, B, N);

    size_t shm1 = (size_t)N * 4 + 256 * 4 + 256 * 4;
    fps_kernel<<<B, 256, shm1, stream>>>(xyz_t, N, 128, fps1);
    gather_xyz_kernel<<<(B * 128 + 255) / 256, 256, 0, stream>>>(
        xyz_t, fps1, l1_xyz, N, 128, B * 128);

    int pi = 1;  // walking d_in: (W, b, g, bt) per layer

    auto launch_layer = [&](const float* Xin, const float* points,
                            const float* xyzp, const float* nxyzp,
                            const int* bq, int Np, int S, int K, int Cpts,
                            float* Yout, int P, int cin, int cout, int ldY,
                            int colOff, int mode) {
        const float* Wp = (const float*)d_in[pi + 0];
        const float* bp = (const float*)d_in[pi + 1];
        const float* gp = (const float*)d_in[pi + 2];
        const float* tp = (const float*)d_in[pi + 3];
        pi += 4;
        pack_w_kernel<<<(cout * cin + 255) / 256, 256, 0, stream>>>(
            Wp, wtBuf, cout, cin);
        if (cout % 64 == 0) {
            dim3 grid(P / 16, cout / 64);
            mlp_gemm_kernel<4><<<grid, 32, 0, stream>>>(
                Xin, points, xyzp, nxyzp, bq, Np, S, K, Cpts,
                wtBuf, bp, gp, tp, Yout, P, cin, cout, ldY, colOff, mode);
        } else {
            dim3 grid(P / 16, cout / 32);
            mlp_gemm_kernel<2><<<grid, 32, 0, stream>>>(
                Xin, points, xyzp, nxyzp, bq, Np, S, K, Cpts,
                wtBuf, bp, gp, tp, Yout, P, cin, cout, ldY, colOff, mode);
        }
    };

    // ---------------- SA1 ----------------
    {
        const int S = 128, Np = N;
        const int   Ks[3] = {4, 8, 16};
        const float rs[3] = {0.1f, 0.2f, 0.4f};
        const int chs[3][3] = {{32, 32, 64}, {64, 64, 128}, {64, 96, 128}};
        const int offs[3] = {0, 64, 192};
        for (int br = 0; br < 3; ++br) {
            int K = Ks[br];
            float r2 = rs[br] * rs[br];
            ball_query_kernel<<<(B * S + 255) / 256, 256, 0, stream>>>(
                xyz_t, l1_xyz, Np, S, K, r2, ballIdx, B * S);
            int P = B * S * K, cin = 3, mode = 1;
            const float* Xin = nullptr;
            float* bufs[2] = {tmp0, tmp1};
            for (int l = 0; l < 3; ++l) {
                int cout = chs[br][l];
                float* Yout = bufs[l & 1];
                launch_layer(Xin, nullptr, xyz_t, l1_xyz, ballIdx, Np, S, K, 0,
                             Yout, P, cin, cout, cout, 0, mode);
                Xin = Yout; cin = cout; mode = 0;
            }
            maxpool_kernel<<<(B * S * chs[br][2] + 255) / 256, 256, 0, stream>>>(
                tmp0, l1_pts, B * S, K, chs[br][2], 320, offs[br]);
        }
    }

    // ---------------- SA2 ----------------
    size_t shm2 = (size_t)128 * 4 + 256 * 4 + 256 * 4;
    fps_kernel<<<B, 256, shm2, stream>>>(l1_xyz, 128, 64, fps2);
    gather_xyz_kernel<<<(B * 64 + 255) / 256, 256, 0, stream>>>(
        l1_xyz, fps2, l2_xyz, 128, 64, B * 64);
    {
        const int S = 64, Np = 128, Cpts = 320;
        const int   Ks[3] = {4, 8, 16};
        const float rs[3] = {0.2f, 0.4f, 0.8f};
        const int chs[3][3] = {{64, 64, 128}, {128, 128, 256}, {128, 128, 256}};
        const int offs[3] = {0, 128, 384};
        for (int br = 0; br < 3; ++br) {
            int K = Ks[br];
            float r2 = rs[br] * rs[br];
            ball_query_kernel<<<(B * S + 255) / 256, 256, 0, stream>>>(
                l1_xyz, l2_xyz, Np, S, K, r2, ballIdx, B * S);
            int P = B * S * K, cin = 323, mode = 1;
            const float* Xin = nullptr;
            float* bufs[2] = {tmp0, tmp1};
            for (int l = 0; l < 3; ++l) {
                int cout = chs[br][l];
                float* Yout = bufs[l & 1];
                launch_layer(Xin, l1_pts, l1_xyz, l2_xyz, ballIdx, Np, S, K,
                             Cpts, Yout, P, cin, cout, cout, 0, mode);
                Xin = Yout; cin = cout; mode = 0;
            }
            maxpool_kernel<<<(B * S * chs[br][2] + 255) / 256, 256, 0, stream>>>(
                tmp0, l2_pts, B * S, K, chs[br][2], 640, offs[br]);
        }
    }

    // ---------------- SA3 ----------------
    concat_sa3_kernel<<<(2048 * 643 + 255) / 256, 256, 0, stream>>>(
        l2_xyz, l2_pts, cat3, 2048);
    {
        const int chs[3] = {256, 512, 1024};
        int P = 2048, cin = 643;
        const float* Xin = cat3;
        float* bufs[2] = {tmp0, tmp1};
        for (int l = 0; l < 3; ++l) {
            int cout = chs[l];
            float* Yout = bufs[l & 1];
            launch_layer(Xin, nullptr, nullptr, nullptr, nullptr, 0, 1, 1, 0,
                         Yout, P, cin, cout, cout, 0, 0);
            Xin = Yout; cin = cout;
        }
        maxpool_kernel<<<(32 * 1024 + 255) / 256, 256, 0, stream>>>(
            tmp0, (float*)d_out, 32, 64, 1024, 1024, 0);
    }
}